// TrafficModel_41669772706063
// MI455X (gfx1250) — compile-verified
//
#include <hip/hip_runtime.h>
#include <hip/hip_bf16.h>

#define L_LINKS 2048
#define P_PATHS 32768
#define O_PAIRS 8192

typedef __attribute__((ext_vector_type(2))) float v2f;
typedef __attribute__((ext_vector_type(8))) float v8f;

// Kernel 1: compute BPR link times t[l], write to out.t; zero out.x (atomic target).
__global__ void bpr_time_kernel(const float* __restrict__ x_hat,
                                const float* __restrict__ alpha,
                                const float* __restrict__ beta,
                                const float* __restrict__ t_min,
                                const float* __restrict__ x_max,
                                float* __restrict__ out) {
    int i = blockIdx.x * blockDim.x + threadIdx.x;
    if (i < L_LINKS) {
        float r  = x_hat[i] / x_max[i];
        float b  = 1.0f + alpha[i] * r;
        float tt = t_min[i] * powf(b, beta[i]);
        out[L_LINKS + i] = tt;   // t output
        out[i] = 0.0f;           // zero x for atomic accumulation
    }
}

// Kernel 2 (fused): u = D^T t via V_WMMA_F32_16X16X4_F32, group-of-4 softmax,
// f = q_hat * p, then x += D * f over this WG's 64-column tile (L2-hot re-read).
// Grid: 512 blocks x 128 threads (4 waves). Each wave owns 16 consecutive paths.
__global__ void traffic_fused_kernel(const float* __restrict__ x_hat,
                                     const float* __restrict__ alpha,
                                     const float* __restrict__ beta,
                                     const float* __restrict__ q_hat,
                                     const float* __restrict__ D,
                                     const float* __restrict__ t_min,
                                     const float* __restrict__ x_max,
                                     float* __restrict__ out) {
    __shared__ float t_s[L_LINKS];   // 8 KB
    __shared__ float f_s[64];        // this WG's 64 path flows

    const int tid  = threadIdx.x;
    const int lane = tid & 31;
    const int wave = tid >> 5;

    // Recompute t into LDS (cheap; keeps kernel self-contained and L2-friendly).
    for (int i = tid; i < L_LINKS; i += 128) {
        float r = x_hat[i] / x_max[i];
        t_s[i] = t_min[i] * powf(1.0f + alpha[i] * r, beta[i]);
    }
    __syncthreads();

    const int p_base = blockIdx.x * 64;        // first path of this WG
    const int p0     = p_base + wave * 16;     // first path of this wave
    const int col    = p0 + (lane & 15);       // path handled by this lane (lanes 0-15)
    const int rsel   = (lane >> 4) << 1;       // lanes 0-15 -> K rows k,k+1; 16-31 -> k+2,k+3
    const bool row0  = (lane & 15) == 0;       // A-matrix row M=0 carriers (lanes 0 and 16)

    // ---- Phase 1: u[col] = sum_k D[k,col] * t[k] via fp32 WMMA ----
    v8f acc = {0.f, 0.f, 0.f, 0.f, 0.f, 0.f, 0.f, 0.f};
    for (int k = 0; k < L_LINKS; k += 4) {
        // A (16x4): row M=0 holds t[k..k+3]; other rows zero.
        float2 tv = *reinterpret_cast<const float2*>(&t_s[k + rsel]);
        v2f a;
        a.x = row0 ? tv.x : 0.0f;
        a.y = row0 ? tv.y : 0.0f;
        // B (4x16): D[k+rsel, col] and D[k+rsel+1, col] (two coalesced 64B segments).
        const float* bp = D + (size_t)(k + rsel) * P_PATHS + col;
        v2f b;
        b.x = bp[0];
        b.y = bp[P_PATHS];
        acc = __builtin_amdgcn_wmma_f32_16x16x4_f32(
            false, a, false, b, (short)0, acc, false, false);
    }
    float u = acc[0];  // lanes 0-15: u for path p0+lane; lanes 16-31: zero row M=8

    // ---- Segmented softmax over groups of 4 lanes (= 4 paths per OD) ----
    float m = fmaxf(u, __shfl_xor(u, 1, 32));
    m = fmaxf(m, __shfl_xor(m, 2, 32));
    float e = __expf(u - m);
    float s = e + __shfl_xor(e, 1, 32);
    s = s + __shfl_xor(s, 2, 32);
    float prob = e / s;

    float f = 0.0f;
    if (lane < 16) {
        f = q_hat[col >> 2] * prob;
        out[4096 + col]  = f;      // f output at offset 2048+2048
        out[36864 + col] = prob;   // p output at offset 4096+32768
        f_s[wave * 16 + lane] = f;
    }
    __syncthreads();

    // ---- Phase 2: x[l] += sum_{j<64} D[l, p_base+j] * f_s[j]  (tile is L2-hot) ----
    const float2* fv = reinterpret_cast<const float2*>(f_s);
    float2 fpair = fv[lane];                        // f_s[2*lane], f_s[2*lane+1]
    const int l_begin = wave * (L_LINKS / 4);
    const int l_end   = l_begin + (L_LINKS / 4);
    for (int l = l_begin; l < l_end; ++l) {
        const float2 dv = *reinterpret_cast<const float2*>(
            D + (size_t)l * P_PATHS + p_base + 2 * lane);   // coalesced 256B row chunk
        float ps = dv.x * fpair.x + dv.y * fpair.y;
        ps += __shfl_xor(ps, 16, 32);
        ps += __shfl_xor(ps,  8, 32);
        ps += __shfl_xor(ps,  4, 32);
        ps += __shfl_xor(ps,  2, 32);
        ps += __shfl_xor(ps,  1, 32);
        if (lane == 0) atomicAdd(&out[l], ps);
    }
}

extern "C" void kernel_launch(void* const* d_in, const int* in_sizes, int n_in,
                              void* d_out, int out_size, void* d_ws, size_t ws_size,
                              hipStream_t stream) {
    const float* x_hat = (const float*)d_in[0];
    const float* alpha = (const float*)d_in[1];
    const float* beta  = (const float*)d_in[2];
    const float* q_hat = (const float*)d_in[3];
    const float* D     = (const float*)d_in[4];
    // d_in[5] = M (one-hot seg = p//4, exploited analytically)
    const float* t_min = (const float*)d_in[6];
    const float* x_max = (const float*)d_in[7];
    float* out = (float*)d_out;  // layout: x[2048] | t[2048] | f[32768] | p[32768]

    bpr_time_kernel<<<8, 256, 0, stream>>>(x_hat, alpha, beta, t_min, x_max, out);
    traffic_fused_kernel<<<P_PATHS / 64, 128, 0, stream>>>(
        x_hat, alpha, beta, q_hat, D, t_min, x_max, out);
}